// Word2VecHSM_936302870888
// MI455X (gfx1250) — compile-verified
//
#include <hip/hip_runtime.h>
#include <hip/hip_bf16.h>
#include <math.h>

// Word2Vec hierarchical-softmax loss on gfx1250 (MI455X, wave32).
// Strategy: 16 samples per wave. Per depth level s, compute the 16x16 GEMM
//   C_s = E (16x256) * W_s^T (256x16)  via 64 chained V_WMMA_F32_16X16X4_F32,
// keep only diag(C_s) (the 16x waste is irrelevant: real work is ~126 MFLOP
// vs a multi-TFLOP f32 WMMA ceiling; the kernel is L2-gather bound, both
// tables fit in the 192MB L2, HBM floor ~3us at 23.3 TB/s).

typedef float v2f __attribute__((ext_vector_type(2)));
typedef float v8f __attribute__((ext_vector_type(8)));

#define HSM_VOCAB 32768
#define HSM_DIM   256
#define HSM_BATCH 16384
#define HSM_TREE  32768
#define HSM_DEPTH 15
#define TILES     (HSM_BATCH / 16)   // 1024 waves total

__global__ __launch_bounds__(256) void
hsm_wmma_kernel(const int* __restrict__ input,
                const int* __restrict__ target,
                const float* __restrict__ emb_table,
                const float* __restrict__ node_W,
                float* __restrict__ partial)
{
    const int lane = threadIdx.x & 31;          // wave32
    const int wave = threadIdx.x >> 5;          // 8 waves per block
    const int tile = blockIdx.x * 8 + wave;     // 16-sample tile
    const int n    = lane & 15;                 // sample-in-tile (rows of A and B)
    const int half = lane >> 4;                 // K-pair select: {0,1} vs {2,3}
    const int b    = tile * 16 + n;

    const int inp  = input[b];
    const int leaf = target[b] + HSM_TREE;

    // A-fragment source: embedding row of this lane's sample, offset by K-pair.
    const float* erow = emb_table + (size_t)inp * HSM_DIM + half * 2;

    // B-fragment sources: 15 tree-node rows along the path to the leaf.
    const float* wrow[HSM_DEPTH];
#pragma unroll
    for (int s = 0; s < HSM_DEPTH; ++s) {
        const int node = leaf >> (HSM_DEPTH - s);       // shifts = 15..1
        wrow[s] = node_W + (size_t)node * HSM_DIM + half * 2;
    }

    v8f acc[HSM_DEPTH];
#pragma unroll
    for (int s = 0; s < HSM_DEPTH; ++s) acc[s] = (v8f){0,0,0,0,0,0,0,0};

    // K loop: one A-chunk load reused by 15 independent WMMA accumulator chains.
    for (int k0 = 0; k0 < HSM_DIM; k0 += 4) {
        v2f a;
        {
            const float2 t2 = *(const float2*)(erow + k0);   // global_load_b64
            a[0] = t2.x; a[1] = t2.y;
        }
#pragma unroll
        for (int s = 0; s < HSM_DEPTH; ++s) {
            v2f bf;
            const float2 t2 = *(const float2*)(wrow[s] + k0); // global_load_b64
            bf[0] = t2.x; bf[1] = t2.y;
            // D = A(16x4 f32) * B(4x16 f32) + C(16x16 f32)
            acc[s] = __builtin_amdgcn_wmma_f32_16x16x4_f32(
                false, a, false, bf, (short)0, acc[s], false, false);
        }
    }

    // Diagonal extraction: C[n,n] lives in lanes 0-7 (vgpr = lane) and
    // lanes 24-31 (vgpr = lane & 7). Other lanes contribute 0.
    const bool isDiag = (lane < 8) || (lane >= 24);
    const int  r      = lane & 7;

    float loss = 0.0f;
#pragma unroll
    for (int s = 0; s < HSM_DEPTH; ++s) {
        float d = acc[s][0];
#pragma unroll
        for (int j = 1; j < 8; ++j) d = (r == j) ? acc[s][j] : d; // v_cndmask chain
        const int dir = (leaf >> (HSM_DEPTH - 1 - s)) & 1;
        const float x = dir ? d : -d;
        // -log_sigmoid(x) = softplus(-x), numerically stable form
        loss += fmaxf(-x, 0.0f) + log1pf(__expf(-fabsf(x)));
    }
    loss = isDiag ? (loss + 1.0f) : 0.0f;   // per_sample = 1 + sum(softplus)

    // Full-wave butterfly reduction (deterministic, fixed order).
#pragma unroll
    for (int off = 16; off > 0; off >>= 1)
        loss += __shfl_xor(loss, off, 32);

    if (lane == 0) partial[tile] = loss;
}

__global__ __launch_bounds__(256) void
hsm_reduce_kernel(const float* __restrict__ partial, float* __restrict__ out)
{
    __shared__ float sm[256];
    float s = 0.0f;
    for (int i = threadIdx.x; i < TILES; i += 256) s += partial[i];
    sm[threadIdx.x] = s;
    __syncthreads();
    for (int stride = 128; stride > 0; stride >>= 1) {
        if ((int)threadIdx.x < stride) sm[threadIdx.x] += sm[threadIdx.x + stride];
        __syncthreads();
    }
    if (threadIdx.x == 0) out[0] = sm[0] * (1.0f / (float)HSM_BATCH);
}

extern "C" void kernel_launch(void* const* d_in, const int* in_sizes, int n_in,
                              void* d_out, int out_size, void* d_ws, size_t ws_size,
                              hipStream_t stream)
{
    const int*   input     = (const int*)d_in[0];
    const int*   target    = (const int*)d_in[1];
    const float* emb_table = (const float*)d_in[2];
    const float* node_W    = (const float*)d_in[3];
    float*       out       = (float*)d_out;
    float*       partial   = (float*)d_ws;          // TILES floats = 4 KB

    // 1024 tiles, 8 waves (of 32) per block -> 128 blocks.
    hsm_wmma_kernel<<<TILES / 8, 256, 0, stream>>>(input, target, emb_table,
                                                   node_W, partial);
    hsm_reduce_kernel<<<1, 256, 0, stream>>>(partial, out);
}